// MultiHeadedRelAttention_69836168233161
// MI455X (gfx1250) — compile-verified
//
#include <hip/hip_runtime.h>
#include <hip/hip_bf16.h>

// ---------------------------------------------------------------------------
// MultiHeadedRelAttention for MI455X (gfx1250).
// B=4, L=256, D=256, H=8, DH=32.
// Bandwidth-bound on the 256MB one-shot read of rel_emb (~11us floor at
// 23.3 TB/s). rel loads + attn stores are non-temporal so the hot K/V
// working set (2MB) stays resident in the 192MB L2 (K/V are logically
// re-read 256MB each across blocks). GEMMs use fp32 WMMA
// (v_wmma_f32_16x16x4_f32); the rel term is a per-(b,q,k) dot streamed
// with 128-bit loads + wave shuffles.
// ---------------------------------------------------------------------------

typedef __attribute__((ext_vector_type(2))) float v2f;
typedef __attribute__((ext_vector_type(4))) float v4f;
typedef __attribute__((ext_vector_type(8))) float v8f;

#define Bb 4
#define Ll 256
#define Dd 256
#define Hh 8
#define DH 32
#define M_TOT (Bb * Ll)   // 1024

// ---------------------------------------------------------------------------
// Kernel 1: Q/K/V projections.  C[m,n] = sum_k X[m,k]*W[n,k] + bias[n]
// One wave per 16x16 output tile; 4 waves / block cover 4 n-tiles.
// grid = (M/16, N/64, 3), block = 128.  z selects (Wq,Wk,Wv).
// WMMA f32 16x16x4 per ISA 7.12.2:
//   A (16x4): lane l -> m=l%16; VGPR0 = K=khalf, VGPR1 = K=khalf+1, khalf=(l/16)*2
//   B (4x16): lane l -> n=l%16; same K striping as A
//   C (16x16): VGPR r -> row r + (l/16)*8, col l%16
// ---------------------------------------------------------------------------
__global__ __launch_bounds__(128)
void qkv_proj_wmma(const float* __restrict__ X,
                   const float* __restrict__ Wq, const float* __restrict__ bq,
                   const float* __restrict__ Wk, const float* __restrict__ bk,
                   const float* __restrict__ Wv, const float* __restrict__ bv,
                   float* __restrict__ Q, float* __restrict__ K,
                   float* __restrict__ V) {
  const int lane = threadIdx.x & 31;
  const int wv   = threadIdx.x >> 5;           // wave in block: 0..3
  const int m0   = blockIdx.x * 16;
  const int n0   = (blockIdx.y * 4 + wv) * 16;
  const int z    = blockIdx.z;

  const float* W    = (z == 0) ? Wq : (z == 1) ? Wk : Wv;
  const float* bias = (z == 0) ? bq : (z == 1) ? bk : bv;
  float*       dst  = (z == 0) ? Q  : (z == 1) ? K  : V;
  const float scale = (z == 0) ? 0.17677669529663687f : 1.0f;  // 1/sqrt(DH)

  const int mrow  = m0 + (lane & 15);
  const int nrow  = n0 + (lane & 15);
  const int khalf = (lane >> 4) << 1;

  v8f acc = {};
  for (int kb = 0; kb < Dd; kb += 4) {
    v2f a, b;
    a.x = X[mrow * Dd + kb + khalf];
    a.y = X[mrow * Dd + kb + khalf + 1];
    b.x = W[nrow * Dd + kb + khalf];
    b.y = W[nrow * Dd + kb + khalf + 1];
    acc = __builtin_amdgcn_wmma_f32_16x16x4_f32(false, a, false, b,
                                                (short)0, acc, false, false);
  }

  const int col = n0 + (lane & 15);
  const int h   = col >> 5;
  const int dh  = col & 31;
  const float bb = bias[col];
#pragma unroll
  for (int r = 0; r < 8; ++r) {
    const int row = m0 + r + ((lane >> 4) << 3);
    const int bi  = row >> 8;        // batch
    const int qp  = row & 255;       // position
    dst[(((bi * Hh + h) * Ll) + qp) * DH + dh] = (acc[r] + bb) * scale;
  }
}

// ---------------------------------------------------------------------------
// Kernel 2: fused scores (content + rel) + mask + softmax + attn@V.
// One 256-thread block per (b,q).  Streams the rel[b,q] slab (256 KB)
// fully coalesced & non-temporal: 8 lanes x float4 = one 128B row per
// (k,h) pair, 32 pairs per step = 4KB contiguous; unrolled x2 for MLP.
// ---------------------------------------------------------------------------
__global__ __launch_bounds__(256)
void attn_kernel(const float* __restrict__ Q, const float* __restrict__ Km,
                 const float* __restrict__ Vm, const float* __restrict__ rel,
                 const unsigned char* __restrict__ mask,
                 const float* __restrict__ u, const float* __restrict__ vparam,
                 float* __restrict__ attn_out, float* __restrict__ ctx) {
  __shared__ float s_qu[Dd];
  __shared__ float s_qv[Dd];
  __shared__ float s_p[Hh][Ll];

  const int t = threadIdx.x;
  const int b = blockIdx.x >> 8;
  const int q = blockIdx.x & 255;

  {  // load this query's (pre-scaled) Q and form q+u, q+v
    const int h = t >> 5, dh = t & 31;
    const float qval = Q[(((b * Hh + h) * Ll) + q) * DH + dh];
    s_qu[t] = qval + u[t];
    s_qv[t] = qval + vparam[t];
  }
  __syncthreads();

  // ---- scores: a_c + b_d, masked -------------------------------------
  const int g = t >> 3;  // group 0..31, one (k,h) pair per group per step
  const int j = t & 7;   // lane within group: covers d = 4j..4j+3
  const v4f* rel4 = (const v4f*)(rel + (size_t)(b * Ll + q) * (Ll * Dd));
  const v4f* Km4  = (const v4f*)Km;
  const unsigned char* mrow = mask + ((size_t)b * Ll + q) * Ll;

#pragma unroll 2
  for (int it = 0; it < 64; ++it) {
    const int pid = it * 32 + g;
    const int k = pid >> 3;
    const int h = pid & 7;
    // rel is a 256MB one-shot stream: non-temporal so it does not sweep
    // K/V (and other hot data) out of the 192MB L2.
    const v4f r4 = __builtin_nontemporal_load(&rel4[(k * Hh + h) * 8 + j]);
    const v4f k4 = Km4[((b * Hh + h) * Ll + k) * 8 + j];
    const float* qv = &s_qv[h * DH + j * 4];
    const float* qu = &s_qu[h * DH + j * 4];
    float p = r4.x * qv[0] + r4.y * qv[1] + r4.z * qv[2] + r4.w * qv[3] +
              k4.x * qu[0] + k4.y * qu[1] + k4.z * qu[2] + k4.w * qu[3];
    p += __shfl_xor(p, 1, 32);
    p += __shfl_xor(p, 2, 32);
    p += __shfl_xor(p, 4, 32);
    if (j == 0) s_p[h][k] = mrow[k] ? -1e18f : p;
  }
  __syncthreads();

  // ---- softmax over k, one wave per head ------------------------------
  {
    const int h = t >> 5, l = t & 31;
    float mx = -3.0e38f;
    float e[8];
#pragma unroll
    for (int i = 0; i < 8; ++i) mx = fmaxf(mx, s_p[h][l + i * 32]);
#pragma unroll
    for (int off = 16; off; off >>= 1) mx = fmaxf(mx, __shfl_xor(mx, off, 32));
    float sum = 0.f;
#pragma unroll
    for (int i = 0; i < 8; ++i) {
      e[i] = __expf(s_p[h][l + i * 32] - mx);
      sum += e[i];
    }
#pragma unroll
    for (int off = 16; off; off >>= 1) sum += __shfl_xor(sum, off, 32);
    const float inv = 1.0f / sum;
    // attn output (8MB) is written once and never re-read on device:
    // non-temporal store keeps it out of L2.
    float* arow = attn_out + (((size_t)(b * Hh + h) * Ll) + q) * Ll;
#pragma unroll
    for (int i = 0; i < 8; ++i) {
      const float a = e[i] * inv;
      s_p[h][l + i * 32] = a;
      __builtin_nontemporal_store(a, &arow[l + i * 32]);
    }
  }
  __syncthreads();

  // ---- ctx[b,q,h,dh] = sum_k attn[h,k] * V[b,h,k,dh] ------------------
  {
    const int h = t >> 5, dh = t & 31;
    const float* vb = Vm + ((size_t)(b * Hh + h) * Ll) * DH + dh;
    float acc = 0.f;
#pragma unroll 4
    for (int k = 0; k < Ll; ++k) acc = fmaf(s_p[h][k], vb[k * DH], acc);
    ctx[(size_t)(b * Ll + q) * Dd + t] = acc;  // already (B,L,D) order
  }
}

// ---------------------------------------------------------------------------
// Kernel 3: out = ctx @ Wo^T + bo, same WMMA scheme as kernel 1.
// ---------------------------------------------------------------------------
__global__ __launch_bounds__(128)
void out_proj_wmma(const float* __restrict__ ctx, const float* __restrict__ Wo,
                   const float* __restrict__ bo, float* __restrict__ out) {
  const int lane = threadIdx.x & 31;
  const int wv   = threadIdx.x >> 5;
  const int m0   = blockIdx.x * 16;
  const int n0   = (blockIdx.y * 4 + wv) * 16;

  const int mrow  = m0 + (lane & 15);
  const int nrow  = n0 + (lane & 15);
  const int khalf = (lane >> 4) << 1;

  v8f acc = {};
  for (int kb = 0; kb < Dd; kb += 4) {
    v2f a, b;
    a.x = ctx[mrow * Dd + kb + khalf];
    a.y = ctx[mrow * Dd + kb + khalf + 1];
    b.x = Wo[nrow * Dd + kb + khalf];
    b.y = Wo[nrow * Dd + kb + khalf + 1];
    acc = __builtin_amdgcn_wmma_f32_16x16x4_f32(false, a, false, b,
                                                (short)0, acc, false, false);
  }

  const int col = n0 + (lane & 15);
  const float bb = bo[col];
#pragma unroll
  for (int r = 0; r < 8; ++r) {
    const int row = m0 + r + ((lane >> 4) << 3);
    out[row * Dd + col] = acc[r] + bb;
  }
}

// ---------------------------------------------------------------------------
extern "C" void kernel_launch(void* const* d_in, const int* in_sizes, int n_in,
                              void* d_out, int out_size, void* d_ws,
                              size_t ws_size, hipStream_t stream) {
  const float*         X    = (const float*)d_in[0];
  const unsigned char* mask = (const unsigned char*)d_in[1];   // bool mask
  const float*         rel  = (const float*)d_in[2];
  const float*         Wq   = (const float*)d_in[3];
  const float*         bq   = (const float*)d_in[4];
  const float*         Wk   = (const float*)d_in[5];
  const float*         bk   = (const float*)d_in[6];
  const float*         Wv   = (const float*)d_in[7];
  const float*         bv   = (const float*)d_in[8];
  const float*         Wo   = (const float*)d_in[9];
  const float*         bo   = (const float*)d_in[10];
  const float*         u    = (const float*)d_in[11];
  const float*         vp   = (const float*)d_in[12];

  float* out  = (float*)d_out;                   // (B,L,D) = 262144 floats
  float* attn = out + Bb * Ll * Dd;              // (B,H,L,L) = 2097152 floats

  float* ws  = (float*)d_ws;
  float* Q   = ws;                               // (B,H,L,DH)
  float* K   = ws + 1 * Bb * Ll * Dd;
  float* V   = ws + 2 * Bb * Ll * Dd;
  float* ctx = ws + 3 * Bb * Ll * Dd;            // (B,L,D)

  qkv_proj_wmma<<<dim3(M_TOT / 16, Dd / 64, 3), 128, 0, stream>>>(
      X, Wq, bq, Wk, bk, Wv, bv, Q, K, V);

  attn_kernel<<<dim3(Bb * Ll), 256, 0, stream>>>(Q, K, V, rel, mask, u, vp,
                                                 attn, ctx);

  out_proj_wmma<<<dim3(M_TOT / 16, Dd / 64), 128, 0, stream>>>(ctx, Wo, bo,
                                                               out);
}